// CTCLoss_20478404067802
// MI455X (gfx1250) — compile-verified
//
#include <hip/hip_runtime.h>

// CTC loss for T=512, B=128, C=1000, L=32 on gfx1250 (MI455X).
// Phase 1 (bandwidth-bound, ~11us floor @ 23.3 TB/s): per-row logsumexp over C
//   + gather of the 33 needed class logits -> compact [T,B,36] buffer (9.4MB).
// Phase 2: CTC forward DP, 1 workgroup per sample, LDS double-buffered alpha.
// Phase 3: mean over B via chained V_WMMA_F32_16X16X4_F32 reductions.

#define T_   512
#define B_   128
#define C_   1000
#define L_   32
#define S_   65    // 2L+1 extended states
#define JP_  36    // padded per-(t,b) stride for the 33 gathered log-probs
#define NEGF (-1e30f)

typedef __attribute__((ext_vector_type(2))) float v2f;
typedef __attribute__((ext_vector_type(4))) float v4f;
typedef __attribute__((ext_vector_type(8))) float v8f;

// ---------------------------------------------------------------------------
// Kernel 1: one wave32 per (t,b) row. Gather 33 label logits first (row lines
// get cached), then stream the 1000-float row with non-temporal loads into
// registers, two-phase logsumexp (max, then sum-exp), butterfly reductions,
// write lp_ext[j] = logit[j] - logsumexp to compact buffer.
// ---------------------------------------------------------------------------
__global__ __launch_bounds__(256) void ctc_lse_gather(
    const float* __restrict__ preds, const int* __restrict__ targets,
    float* __restrict__ lpbuf) {
  const int lane = threadIdx.x & 31;
  const int row  = blockIdx.x * 8 + (threadIdx.x >> 5);   // row = t*B + b
  const int b    = row & (B_ - 1);
  const float* rowp = preds + (size_t)row * C_;

  // Gather the logits we will actually need (j=0 blank, j=1..32 labels).
  // Done with regular (cacheable) loads before the NT streaming pass.
  const int lab   = (lane == 0) ? 0 : targets[b * L_ + (lane - 1)];
  const float raw = rowp[lab];
  float raw32 = 0.0f;
  if (lane == 0) raw32 = rowp[targets[b * L_ + (L_ - 1)]];

  // Stream the row: 250 float4s, 8 per lane, non-temporal (262MB > L2).
  const v4f* rp = (const v4f*)rowp;
  const float NINF = -__builtin_inff();
  v4f v[8];
#pragma unroll
  for (int j = 0; j < 8; ++j) {
    const int idx = j * 32 + lane;
    if (idx < 250) v[j] = __builtin_nontemporal_load(rp + idx);
    else           v[j] = (v4f){NINF, NINF, NINF, NINF};
  }
  float m = NINF;
#pragma unroll
  for (int j = 0; j < 8; ++j)
    m = fmaxf(m, fmaxf(fmaxf(v[j].x, v[j].y), fmaxf(v[j].z, v[j].w)));
#pragma unroll
  for (int off = 16; off >= 1; off >>= 1)
    m = fmaxf(m, __shfl_xor(m, off, 32));
  float s = 0.0f;
#pragma unroll
  for (int j = 0; j < 8; ++j)
    s += __expf(v[j].x - m) + __expf(v[j].y - m) +
         __expf(v[j].z - m) + __expf(v[j].w - m);
#pragma unroll
  for (int off = 16; off >= 1; off >>= 1)
    s += __shfl_xor(s, off, 32);
  const float Z = m + logf(s);   // logsumexp of the row

  float* outp = lpbuf + (size_t)row * JP_;
  outp[lane] = raw - Z;
  if (lane == 0) outp[32] = raw32 - Z;
}

// ---------------------------------------------------------------------------
// Kernel 2: CTC forward DP. One workgroup (96 threads) per batch sample.
// alpha double-buffered in LDS with 2 NEG pad slots so s-1/s-2 are branchless.
// Logit row for t+1 is prefetched into registers while t is computed.
// ---------------------------------------------------------------------------
__global__ __launch_bounds__(96) void ctc_dp(
    const float* __restrict__ lpbuf, const int* __restrict__ targets,
    const int* __restrict__ preds_lengths, const int* __restrict__ targets_lengths,
    float* __restrict__ nll_out) {
  const int b = blockIdx.x;
  const int s = threadIdx.x;

  __shared__ int   extsh[S_];
  __shared__ int   jmap[S_];
  __shared__ int   skipf[S_];
  __shared__ float A[2][S_ + 2];

  if (s < S_) {
    const int e = (s & 1) ? targets[b * L_ + ((s - 1) >> 1)] : 0;
    extsh[s] = e;
    jmap[s]  = (s & 1) ? (((s - 1) >> 1) + 1) : 0;
  }
  if (s < 2) { A[0][s] = NEGF; A[1][s] = NEGF; }
  __syncthreads();
  if (s < S_) {
    int sk = 0;
    if (s >= 2) { const int e = extsh[s]; sk = (e != 0) && (e != extsh[s - 2]); }
    skipf[s] = sk;
  }
  const int tEnd = preds_lengths[b];
  if (s < S_) {
    float a0 = NEGF;
    if (s < 2) a0 = lpbuf[(size_t)b * JP_ + jmap[s]];
    A[0][s + 2] = a0;
  }
  __syncthreads();

  int cur = 0;
  float pn = 0.0f;
  if (tEnd > 1 && s < S_) pn = lpbuf[((size_t)1 * B_ + b) * JP_ + jmap[s]];
  for (int t = 1; t < tEnd; ++t) {
    const float p = pn;
    if (t + 1 < tEnd && s < S_)
      pn = lpbuf[((size_t)(t + 1) * B_ + b) * JP_ + jmap[s]];   // prefetch
    float nv = NEGF;
    if (s < S_) {
      const float a1 = A[cur][s + 2];
      const float a2 = A[cur][s + 1];
      const float a3 = skipf[s] ? A[cur][s] : NEGF;
      const float mm = fmaxf(a1, fmaxf(a2, a3));
      nv = mm + __logf(__expf(a1 - mm) + __expf(a2 - mm) + __expf(a3 - mm)) + p;
      A[cur ^ 1][s + 2] = nv;
    }
    __syncthreads();   // orders writes(t)->reads(t+1) and reads(t)->writes(t+1)
    cur ^= 1;
  }

  if (s == 0) {
    const int Lb = targets_lengths[b];
    const float e1 = A[cur][2 * Lb - 1 + 2];
    const float e2 = A[cur][2 * Lb + 2];
    const float mm = fmaxf(e1, e2);
    float nll = -(mm + __logf(__expf(e1 - mm) + __expf(e2 - mm)));
    if (!(nll < 1e29f)) nll = 0.0f;   // zero_infinity
    nll_out[b] = nll / (float)Lb;
  }
}

// ---------------------------------------------------------------------------
// Kernel 3: mean over B=128 values via the matrix pipe.
// Two chained V_WMMA_F32_16X16X4_F32: A = 16x4 value chunk, B = ones 4x16,
// accumulate into C. D[m,n] == rowsum(m) for every n; each lane sums its 8
// accumulator VGPRs (rows 0-7 / 8-15), combine halves with xor-16 shuffle.
// Single wave32, no divergence before the WMMAs (EXEC must be all ones).
// ---------------------------------------------------------------------------
__global__ __launch_bounds__(32) void ctc_mean_wmma(
    const float* __restrict__ nll, float* __restrict__ out) {
  const int l = threadIdx.x;
  // A 16x4 layout: lanes 0-15 hold K=0,1 of row M=lane; lanes 16-31 K=2,3.
  const int base = (l < 16) ? (l * 4) : ((l - 16) * 4 + 2);
  v2f a0, a1, bo;
  a0.x = nll[base];          a0.y = nll[base + 1];
  a1.x = nll[64 + base];     a1.y = nll[64 + base + 1];
  bo.x = 1.0f;               bo.y = 1.0f;   // B = all-ones (layout-agnostic)
  v8f c = {};
  c = __builtin_amdgcn_wmma_f32_16x16x4_f32(false, a0, false, bo, (short)0, c,
                                            false, false);
  c = __builtin_amdgcn_wmma_f32_16x16x4_f32(false, a1, false, bo, (short)0, c,
                                            false, false);
  float loc = c[0] + c[1] + c[2] + c[3] + c[4] + c[5] + c[6] + c[7];
  loc += __shfl_xor(loc, 16, 32);   // rows 0-7 + rows 8-15
  if (l == 0) out[0] = loc * (1.0f / (float)B_);
}

// ---------------------------------------------------------------------------
extern "C" void kernel_launch(void* const* d_in, const int* in_sizes, int n_in,
                              void* d_out, int out_size, void* d_ws, size_t ws_size,
                              hipStream_t stream) {
  (void)in_sizes; (void)n_in; (void)out_size; (void)ws_size;
  const float* preds           = (const float*)d_in[0];
  const int*   targets         = (const int*)d_in[1];
  const int*   preds_lengths   = (const int*)d_in[2];
  const int*   targets_lengths = (const int*)d_in[3];

  float* lpbuf  = (float*)d_ws;                          // T*B*JP_ floats (~9.4MB)
  float* nllbuf = lpbuf + (size_t)T_ * B_ * JP_;         // B floats

  ctc_lse_gather<<<(T_ * B_) / 8, 256, 0, stream>>>(preds, targets, lpbuf);
  ctc_dp<<<B_, 96, 0, stream>>>(lpbuf, targets, preds_lengths, targets_lengths,
                                nllbuf);
  ctc_mean_wmma<<<1, 32, 0, stream>>>(nllbuf, (float*)d_out);
}